// PreProcessCGCNNLayer_74156905332878
// MI455X (gfx1250) — compile-verified
//
#include <hip/hip_runtime.h>

// Problem constants (match reference)
#define NUM_ATOM 4096
#define NUM_NBR  32
#define BATCH    8
#define NG       41                                   // len(arange(0, 8.0+0.2, 0.2))
#define PAIRS_PER_CH (BATCH * NUM_ATOM * NUM_NBR)     // 1,048,576
#define TPB      256
#define BLOCKS_PER_CH (PAIRS_PER_CH / TPB)            // 4096
#define FLOATS_PER_BLOCK (TPB * NG)                   // 10,496 (divisible by 4)
#define VEC4_PER_BLOCK (FLOATS_PER_BLOCK / 4)         // 2,624

// -25 * log2(e)  ==  -1/VAR^2 * log2(e), VAR = 0.2
#define NEG_INVVAR2_LOG2E (-36.067376022224085f)

typedef float v4f __attribute__((ext_vector_type(4)));

__global__ __launch_bounds__(TPB)
void bond_fea_kernel(const float* __restrict__ coords,   // (B, 4096, 3, 2)
                     const float* __restrict__ lat,      // (B, 3, 2)
                     const int*   __restrict__ nbr,      // (B, 4096, 32, 2)
                     float*       __restrict__ out)
{
    __shared__ float s_dist[TPB];

    const int tid = threadIdx.x;
    const int blk = blockIdx.x;
    const int c   = (blk >= BLOCKS_PER_CH) ? 1 : 0;     // channel (slow grid dim)
    const int cb  = blk - c * BLOCKS_PER_CH;            // block index within channel
    const int pc  = cb * TPB + tid;                     // pair index within channel
                                                        // pc == (b*4096 + i)*32 + j

    // ---------------- Phase 1: one distance per thread ----------------
    const int b   = pc >> 17;                           // / (4096*32); b const per block
    const int rem = pc & (NUM_ATOM * NUM_NBR - 1);
    const int i   = rem >> 5;

    const int n = nbr[pc * 2 + c];                      // neighbor atom index

    const int selfBase = (b * NUM_ATOM + i) * 6 + c;    // interleaved last dim of 2
    const int nbrBase  = (b * NUM_ATOM + n) * 6 + c;
    const int latBase  = b * 6 + c;

    float dist2 = 0.0f;
#pragma unroll
    for (int d = 0; d < 3; ++d) {
        const float L  = lat[latBase + d * 2];
        float dd = fabsf(coords[selfBase + d * 2] - coords[nbrBase + d * 2]);
        dd = (dd > 0.5f * L) ? (dd - L) : dd;           // minimum image
        dist2 = fmaf(dd, dd, dist2);
    }
    s_dist[tid] = sqrtf(dist2);

    // nbr output (reference returns int32; d_out is a float buffer -> exact cast)
    const long nbrSeg = c ? (long)42 * PAIRS_PER_CH : 0L;
    out[nbrSeg + pc] = (float)n;

    __syncthreads();

    // ---------------- Phase 2: coalesced b128 NT stores of 41 bins/pair ----------------
    const long bondSeg = c ? (long)43 * PAIRS_PER_CH : (long)PAIRS_PER_CH;
    float* __restrict__ dst = out + bondSeg + (long)cb * FLOATS_PER_BLOCK;

    for (int v = tid; v < VEC4_PER_BLOCK; v += TPB) {
        const int base = v * 4;
        v4f r;
#pragma unroll
        for (int k = 0; k < 4; ++k) {
            const int   e  = base + k;
            const int   lp = e / NG;                    // magic-number division
            const int   g  = e - lp * NG;
            const float d  = s_dist[lp] - 0.2f * (float)g;
            r[k] = __builtin_amdgcn_exp2f(d * d * NEG_INVVAR2_LOG2E); // v_exp_f32
        }
        __builtin_nontemporal_store(r, (v4f*)(dst + base));          // 128-bit NT store
    }
}

extern "C" void kernel_launch(void* const* d_in, const int* in_sizes, int n_in,
                              void* d_out, int out_size, void* d_ws, size_t ws_size,
                              hipStream_t stream)
{
    const float* coords = (const float*)d_in[0];  // (B, 4096, 3, 2)  f32
    const float* lat    = (const float*)d_in[1];  // (B, 3, 2)        f32
    const int*   nbr    = (const int*)  d_in[2];  // (B, 4096, 32, 2) i32
    float* out = (float*)d_out;                   // [nbr1 | fea1 | nbr2 | fea2]

    bond_fea_kernel<<<dim3(2 * BLOCKS_PER_CH), dim3(TPB), 0, stream>>>(coords, lat, nbr, out);
}